// PWRSWtL_33328946217516
// MI455X (gfx1250) — compile-verified
//
#include <hip/hip_runtime.h>

typedef __attribute__((ext_vector_type(2))) float v2f;
typedef __attribute__((ext_vector_type(8))) float v8f;

#define BINS 256
// N = 64 * 1 * 2048 * 256
#define NTOT 33554432
// N * batch_size(=64) = 2^31
#define DENOM 2147483648.0f

// ---------------------------------------------------------------------------
// Kernel 0: zero the histogram counts and the output accumulator.
// (d_ws / d_out are poisoned with 0xAA and never re-poisoned between replays.)
// ---------------------------------------------------------------------------
__global__ void init_kernel(unsigned int* __restrict__ counts,
                            float* __restrict__ out) {
    int t = threadIdx.x;
    if (t < BINS) counts[t] = 0u;
    if (t == 0)  out[0] = 0.0f;
}

// ---------------------------------------------------------------------------
// Kernel 1: histogram of tar.  LDS-privatized per block, one global merge.
// bin = clip(floor(t * 256/255), 0, 255)
// ---------------------------------------------------------------------------
__global__ void hist_kernel(const float4* __restrict__ tar4,
                            unsigned int* __restrict__ gcounts,
                            int n4) {
    __shared__ unsigned int lhist[BINS];
    for (int i = threadIdx.x; i < BINS; i += blockDim.x) lhist[i] = 0u;
    __syncthreads();

    const float sc = 256.0f / 255.0f;
    int stride = gridDim.x * blockDim.x;
    for (int i = blockIdx.x * blockDim.x + threadIdx.x; i < n4; i += stride) {
        float4 t = tar4[i];
        int b0 = (int)floorf(t.x * sc);
        int b1 = (int)floorf(t.y * sc);
        int b2 = (int)floorf(t.z * sc);
        int b3 = (int)floorf(t.w * sc);
        b0 = min(max(b0, 0), BINS - 1);
        b1 = min(max(b1, 0), BINS - 1);
        b2 = min(max(b2, 0), BINS - 1);
        b3 = min(max(b3, 0), BINS - 1);
        atomicAdd(&lhist[b0], 1u);
        atomicAdd(&lhist[b1], 1u);
        atomicAdd(&lhist[b2], 1u);
        atomicAdd(&lhist[b3], 1u);
    }
    __syncthreads();
    for (int i = threadIdx.x; i < BINS; i += blockDim.x) {
        unsigned int c = lhist[i];
        if (c) atomicAdd(&gcounts[i], c);
    }
}

// ---------------------------------------------------------------------------
// Kernel 2: weights.  One block of 256 threads.
// p = counts/2^31 ; w = 1/(p+1e-12) ; weight = w / sum(w) / N   (1/N folded in)
// ---------------------------------------------------------------------------
__global__ void weight_kernel(const unsigned int* __restrict__ counts,
                              float* __restrict__ weight) {
    __shared__ float s[BINS];
    int t = threadIdx.x;
    float p = (float)counts[t] * (1.0f / DENOM);
    float w = 1.0f / (p + 1e-12f);
    s[t] = w;
    __syncthreads();
    for (int off = BINS / 2; off > 0; off >>= 1) {
        if (t < off) s[t] += s[t + off];
        __syncthreads();
    }
    float wsum = s[0];
    weight[t] = (w / wsum) * (1.0f / (float)NTOT);
}

// ---------------------------------------------------------------------------
// Kernel 3: fused weighted-SSD reduction.
//   acc += weight[idx & 255] * (src - tar)^2
// Wave reduction via V_WMMA_F32_16X16X4_F32:
//   A(16x4): lanes 0-15 carry K=0 (=acc), lanes 16-31 carry K=2 (=acc),
//            K=1/K=3 zeroed.  B = ones  ->  D[m][n] = acc[m] + acc[m+16].
//   Sum of the 8 D VGPRs per lane + shfl_xor(16) = full 32-lane total.
// EXEC is all-1s here: trip counts are uniform (N divisible by grid size),
// no early returns.
// ---------------------------------------------------------------------------
__global__ void loss_kernel(const float4* __restrict__ src4,
                            const float4* __restrict__ tar4,
                            const float* __restrict__ weight,
                            float* __restrict__ out,
                            int n4) {
    __shared__ float wsh[BINS];
    __shared__ float wave_sums[8];
    for (int i = threadIdx.x; i < BINS; i += blockDim.x) wsh[i] = weight[i];
    __syncthreads();

    float acc = 0.0f;
    int stride = gridDim.x * blockDim.x;
    for (int i = blockIdx.x * blockDim.x + threadIdx.x; i < n4; i += stride) {
        float4 s = src4[i];
        float4 t = tar4[i];
        int c0 = (i * 4) & (BINS - 1);   // 4-aligned, never crosses the 256 wrap
        float d0 = s.x - t.x;
        float d1 = s.y - t.y;
        float d2 = s.z - t.z;
        float d3 = s.w - t.w;
        acc += wsh[c0 + 0] * d0 * d0;
        acc += wsh[c0 + 1] * d1 * d1;
        acc += wsh[c0 + 2] * d2 * d2;
        acc += wsh[c0 + 3] * d3 * d3;
    }

    // ---- wave32 reduction via WMMA f32 16x16x4 ----
    v2f a; a.x = acc;  a.y = 0.0f;     // A: K=0 / K=2 slots carry partials
    v2f b; b.x = 1.0f; b.y = 1.0f;     // B: all ones
    v8f c = {};
    v8f d = __builtin_amdgcn_wmma_f32_16x16x4_f32(
        /*neg_a=*/false, a, /*neg_b=*/false, b,
        /*c_mod=*/(short)0, c, /*reuse_a=*/false, /*reuse_b=*/false);
    float half_sum = d[0] + d[1] + d[2] + d[3] + d[4] + d[5] + d[6] + d[7];
    float wave_total = half_sum + __shfl_xor(half_sum, 16, 32);

    int lane = threadIdx.x & 31;
    int wave = threadIdx.x >> 5;
    if (lane == 0) wave_sums[wave] = wave_total;
    __syncthreads();

    if (threadIdx.x == 0) {
        int nwaves = (blockDim.x + 31) >> 5;
        float bsum = 0.0f;
        for (int w = 0; w < nwaves; ++w) bsum += wave_sums[w];
        atomicAdd(out, bsum);
    }
}

// ---------------------------------------------------------------------------
extern "C" void kernel_launch(void* const* d_in, const int* in_sizes, int n_in,
                              void* d_out, int out_size, void* d_ws, size_t ws_size,
                              hipStream_t stream) {
    const float4* src4 = (const float4*)d_in[0];
    const float4* tar4 = (const float4*)d_in[1];
    float* out = (float*)d_out;

    unsigned int* counts = (unsigned int*)d_ws;             // 256 * u32
    float* weight = (float*)((char*)d_ws + BINS * sizeof(unsigned int)); // 256 * f32

    int n4 = NTOT / 4;  // 8,388,608 float4 elements

    init_kernel<<<1, 256, 0, stream>>>(counts, out);
    hist_kernel<<<1024, 256, 0, stream>>>(tar4, counts, n4);
    weight_kernel<<<1, BINS, 0, stream>>>(counts, weight);
    loss_kernel<<<2048, 256, 0, stream>>>(src4, tar4, weight, out, n4);
}